// EMBLossCriterion_26250840113737
// MI455X (gfx1250) — compile-verified
//
#include <hip/hip_runtime.h>
#include <hip/hip_bf16.h>
#include <math.h>

// ---------------------------------------------------------------------------
// EMB loss:  sum_{b,l} gmask[b,l] * sum_n | ||bow_n - gold_l + eps||_2 - pred |
// Core math rewritten as a batched GEMM (gold x bow^T) on the fp32 WMMA pipe:
//   ||mb*b - mg*g + eps||^2 = ||mg g||^2 + ||mb b||^2 - 2 mg mb (g.b)
//                             + 2eps(mb sum_b - mg sum_g) + E eps^2
// Masks are scalars per row/col, so they are hoisted OUT of the K-loop and
// applied once per C element in the epilogue -> inner loop is pure
// 2x global_load_b64 + 1x v_wmma_f32_16x16x4_f32 (max VMEM/WMMA co-exec).
// Problem is memory/gather bound (~0.21 GFLOP vs ~60MB traffic @ 23.3 TB/s),
// so FP32 WMMA is the right precision: exact vs fp32 reference, matrix pipe.
// ---------------------------------------------------------------------------

typedef __attribute__((ext_vector_type(2))) float v2f;
typedef __attribute__((ext_vector_type(8))) float v8f;

#define EPSV 1e-6f

constexpr int B_ = 8, L_ = 128, N_ = 200, V_ = 32000, E_ = 512;
constexpr int TILES_M = L_ / 16;                  // 8
constexpr int TILES_N = (N_ + 15) / 16;           // 13
constexpr int TILES_PER_BATCH = TILES_M * TILES_N;// 104
constexpr int TOTAL_TILES = B_ * TILES_PER_BATCH; // 832
constexpr int GOLD_ROWS = B_ * L_;                // 1024
constexpr int BOW_ROWS  = B_ * N_;                // 1600

// workspace layout (in floats)
constexpr int WS_SUMG = 0;                        // mask * sum(W row)
constexpr int WS_NG2  = WS_SUMG + GOLD_ROWS;      // mask^2 * ||W row||^2
constexpr int WS_SUMB = WS_NG2  + GOLD_ROWS;
constexpr int WS_NB2  = WS_SUMB + BOW_ROWS;
constexpr int WS_PART = WS_NB2  + BOW_ROWS;       // + TOTAL_TILES floats

// ---------------------------------------------------------------------------
// Kernel 1: per-embedding-row masked sum and squared-norm.
// One wave32 per row; rows [0,1024) = gold, [1024,2624) = bow.
// ---------------------------------------------------------------------------
__global__ __launch_bounds__(256) void emb_aux_kernel(
    const int* __restrict__ gold, const float* __restrict__ gmask,
    const int* __restrict__ bow,  const float* __restrict__ bmask,
    const float* __restrict__ W,  float* __restrict__ ws)
{
    const int wave = (blockIdx.x * blockDim.x + threadIdx.x) >> 5;
    const int lane = threadIdx.x & 31;

    const bool isGold = wave < GOLD_ROWS;
    const int  r      = isGold ? wave : (wave - GOLD_ROWS);
    const int  tok    = isGold ? gold[r] : bow[r];
    const float m     = isGold ? gmask[r] : bmask[r];

    const float* __restrict__ row = W + (long)tok * E_;
    float s = 0.0f, q = 0.0f;
    #pragma unroll 4
    for (int e = lane; e < E_; e += 32) {
        float v = row[e];
        s += v;
        q += v * v;
    }
    #pragma unroll
    for (int off = 16; off; off >>= 1) {
        s += __shfl_xor(s, off, 32);
        q += __shfl_xor(q, off, 32);
    }
    if (lane == 0) {
        if (isGold) { ws[WS_SUMG + r] = m * s; ws[WS_NG2 + r] = m * m * q; }
        else        { ws[WS_SUMB + r] = m * s; ws[WS_NB2 + r] = m * m * q; }
    }
}

// ---------------------------------------------------------------------------
// Kernel 2: one wave32 per 16x16 (L,N) tile per batch.
// A = raw gold rows (16xK), B = raw bow rows as KxN, K stepped by 4
// with v_wmma_f32_16x16x4_f32 over E=512 (128 WMMAs per tile).
//
// fp32 16x4 A fragment layout: lanes 0-15 hold row M=lane {K,K+1}, lanes
// 16-31 hold row M=lane-16 {K+2,K+3} -> one aligned 8-byte load per lane.
// B (4x16) is the mirror: lane's column n = lane&15, two consecutive K.
// C/D: vgpr j, lane -> (M = j + 8*(lane>>4), N = lane&15).
// Masks applied per C element in the epilogue (scalars factor out of dot).
// ---------------------------------------------------------------------------
__global__ __launch_bounds__(256) void emb_tile_kernel(
    const float* __restrict__ prob,
    const int* __restrict__ gold, const float* __restrict__ gmask,
    const int* __restrict__ bow,  const float* __restrict__ bmask,
    const float* __restrict__ W,  float* __restrict__ ws)
{
    const int tile = blockIdx.x * (blockDim.x >> 5) + (threadIdx.x >> 5);
    const int lane = threadIdx.x & 31;

    const int b  = tile / TILES_PER_BATCH;
    const int t  = tile % TILES_PER_BATCH;
    const int m0 = (t / TILES_N) * 16;
    const int n0 = (t % TILES_N) * 16;

    const int half = lane >> 4;   // 0: K,K+1   1: K+2,K+3
    const int lo   = lane & 15;

    // ---- A-side (gold) per-lane setup ----
    const int  mA    = m0 + lo;
    const long aBase = (long)gold[b * L_ + mA] * E_ + 2 * half;

    // ---- B-side (bow) per-lane setup (tail n>=200 excluded in epilogue) ----
    const int  n     = n0 + lo;
    const bool valid = (n < N_);
    const int  bCol  = b * N_ + (valid ? n : 0);
    const int  bTok  = valid ? bow[bCol] : 0;
    const float bmv  = valid ? bmask[bCol] : 0.0f;
    const long bBase = (long)bTok * E_ + 2 * half;

    // ---- K loop: pure loads + 128 x v_wmma_f32_16x16x4_f32 ----
    v8f c = {};
    #pragma unroll 4
    for (int k = 0; k < E_; k += 4) {
        v2f a  = *(const v2f*)(W + aBase + k);
        v2f bb = *(const v2f*)(W + bBase + k);
        c = __builtin_amdgcn_wmma_f32_16x16x4_f32(
                /*neg_a=*/false, a, /*neg_b=*/false, bb,
                /*c_mod=*/(short)0, c, /*reuse_a=*/false, /*reuse_b=*/false);
    }

    // ---- epilogue: masks, distances, prob gather, masked |dist - pred| ----
    const float two_eps = 2.0f * EPSV;
    const float ceps    = (float)E_ * EPSV * EPSV;
    float sb = 0.0f, nb2 = 0.0f;
    if (valid) { sb = ws[WS_SUMB + bCol]; nb2 = ws[WS_NB2 + bCol]; }

    float acc = 0.0f;
    #pragma unroll
    for (int j = 0; j < 8; ++j) {
        const int m    = m0 + j + 8 * half;
        const int gRow = b * L_ + m;
        const float sg  = ws[WS_SUMG + gRow];   // mask-scaled
        const float ng2 = ws[WS_NG2  + gRow];   // mask^2-scaled
        const float gm  = gmask[gRow];
        const float dot = gm * bmv * c[j];      // hoisted mask application
        float dist2 = ng2 + nb2 - 2.0f * dot + two_eps * (sb - sg) + ceps;
        float dist  = sqrtf(fmaxf(dist2, 0.0f));
        float pred  = prob[(long)gRow * V_ + bTok];
        float term  = fabsf(dist - pred) * gm;  // outer gold_mask factor
        acc += valid ? term : 0.0f;
    }

    #pragma unroll
    for (int off = 16; off; off >>= 1) acc += __shfl_xor(acc, off, 32);
    if (lane == 0) ws[WS_PART + tile] = acc;
}

// ---------------------------------------------------------------------------
// Kernel 3: deterministic single-block reduction of the 832 tile partials.
// ---------------------------------------------------------------------------
__global__ __launch_bounds__(256) void emb_reduce_kernel(
    const float* __restrict__ ws, float* __restrict__ out)
{
    __shared__ float sm[256];
    float s = 0.0f;
    for (int i = threadIdx.x; i < TOTAL_TILES; i += 256) s += ws[WS_PART + i];
    sm[threadIdx.x] = s;
    __syncthreads();
    #pragma unroll
    for (int off = 128; off; off >>= 1) {
        if ((int)threadIdx.x < off) sm[threadIdx.x] += sm[threadIdx.x + off];
        __syncthreads();
    }
    if (threadIdx.x == 0) out[0] = sm[0];
}

// ---------------------------------------------------------------------------
extern "C" void kernel_launch(void* const* d_in, const int* in_sizes, int n_in,
                              void* d_out, int out_size, void* d_ws, size_t ws_size,
                              hipStream_t stream) {
    (void)in_sizes; (void)n_in; (void)out_size; (void)ws_size;
    const float* prob  = (const float*)d_in[0];  // (B,L,V)
    const int*   gold  = (const int*)  d_in[1];  // (B,L)
    const float* gmask = (const float*)d_in[2];  // (B,L)
    const int*   bow   = (const int*)  d_in[3];  // (B,N)
    const float* bmask = (const float*)d_in[4];  // (B,N)
    const float* W     = (const float*)d_in[5];  // (V,E)
    float* out = (float*)d_out;
    float* ws  = (float*)d_ws;

    // 2624 rows, 8 waves per 256-thread block -> 328 blocks
    emb_aux_kernel<<<(GOLD_ROWS + BOW_ROWS) / 8, 256, 0, stream>>>(
        gold, gmask, bow, bmask, W, ws);

    // 832 tiles, 8 waves per block -> 104 blocks
    emb_tile_kernel<<<TOTAL_TILES / 8, 256, 0, stream>>>(
        prob, gold, gmask, bow, bmask, W, ws);

    emb_reduce_kernel<<<1, 256, 0, stream>>>(ws, out);
}